// SpatialAttention_7258494730247
// MI455X (gfx1250) — compile-verified
//
#include <hip/hip_runtime.h>

// SpatialAttention: mean/max over C, 7x7 conv(2->1, pad 3), sigmoid.
// Shapes fixed by the reference: x[32,256,112,112] f32, W[1,2,7,7] f32.
#define B_   32
#define C_   256
#define H_   112
#define W_   112
#define HW   (H_ * W_)     // 12544
#define HW4  (HW / 4)      // 3136
#define KS   7
#define PAD_ 3
#define TILE 16
#define TW   (TILE + KS - 1)  // 22
#define POOL_BLK 224          // 3136 = 14 * 224 (7 waves/block)

typedef float v4f __attribute__((ext_vector_type(4)));

// ---- gfx1250 async global->LDS path (guarded; falls back to plain copies) ----
#if defined(__has_builtin)
#  if __has_builtin(__builtin_amdgcn_global_load_async_to_lds_b32)
#    define USE_ASYNC_LDS 1
#  endif
#endif
#ifndef USE_ASYNC_LDS
#  define USE_ASYNC_LDS 0
#endif

#if USE_ASYNC_LDS && defined(__has_builtin)
#  if __has_builtin(__builtin_amdgcn_s_wait_asynccnt)
#    define WAIT_ASYNC() __builtin_amdgcn_s_wait_asynccnt(0)
#  else
#    define WAIT_ASYNC() asm volatile("s_wait_asynccnt 0" ::: "memory")
#  endif
#else
#  define WAIT_ASYNC()
#endif

typedef __attribute__((address_space(1))) int gint_t;   // global
typedef __attribute__((address_space(3))) int sint_t;   // LDS

__device__ __forceinline__ v4f vmax4(v4f a, v4f b) {
  a.x = fmaxf(a.x, b.x);
  a.y = fmaxf(a.y, b.y);
  a.z = fmaxf(a.z, b.z);
  a.w = fmaxf(a.w, b.w);
  return a;
}

// ------------------------------------------------------------------
// Stage 1: partial channel sum + max over a CP=C/SPLIT channel slice.
// grid = (14, B, SPLIT), block = 224. Thread -> (b, part, 4 pixels).
// Dual accumulator streams + unroll 4 => 8 NT b128 loads in flight per
// wave body; SPLIT=2 doubles wave count for HBM latency hiding.
// Stores RAW sum (mean scaling folded into stage 2) and max.
// Partial layout: part[(p)][b][2][HW].
// ------------------------------------------------------------------
template <int SPLIT>
__global__ __launch_bounds__(POOL_BLK) void sa_pool(const v4f* __restrict__ x,
                                                    v4f* __restrict__ partial) {
  const int s4   = blockIdx.x * POOL_BLK + threadIdx.x;  // < 3136
  const int b    = blockIdx.y;
  const int part = blockIdx.z;
  constexpr int CP = C_ / SPLIT;                         // channels per part

  const v4f* p = x + (size_t)b * ((size_t)C_ * HW4)
                   + (size_t)part * ((size_t)CP * HW4) + s4;

  v4f sa = {0.0f, 0.0f, 0.0f, 0.0f}, sb = sa;
  const float ninf = -__builtin_inff();
  v4f ma = {ninf, ninf, ninf, ninf}, mb = ma;

#pragma unroll 4
  for (int c = 0; c < CP; c += 2) {
    v4f va = __builtin_nontemporal_load(p);
    v4f vb = __builtin_nontemporal_load(p + HW4);
    p += 2 * HW4;
    sa += va;  ma = vmax4(ma, va);
    sb += vb;  mb = vmax4(mb, vb);
  }
  const v4f sum = sa + sb;
  const v4f mx  = vmax4(ma, mb);

  v4f* dst = partial + (size_t)part * (B_ * 2 * HW4) + (size_t)b * (2 * HW4) + s4;
  dst[0]   = sum;   // raw sum (channel 0)
  dst[HW4] = mx;    // max     (channel 1)
}

// ------------------------------------------------------------------
// Stage 2: combine SPLIT partials, 7x7 conv (2ch->1ch, pad 3) + sigmoid.
// 16x16 output tile per block; SPLIT halo tiles (22x22x2) staged into LDS
// via gfx1250 async global->LDS loads (ASYNCcnt), combined in LDS, then
// fully unrolled 98-tap FMA.
// ------------------------------------------------------------------
template <int SPLIT>
__global__ __launch_bounds__(256) void sa_conv(const float* __restrict__ partial,
                                               const float* __restrict__ w,
                                               float* __restrict__ out) {
  __shared__ float tile[SPLIT * 2 * TW * TW];   // <= 2*968 floats
  __shared__ float wsh[2 * KS * KS];            // 98 floats

  const int tid = threadIdx.x;
  const int bx = blockIdx.x % (W_ / TILE);
  const int by = (blockIdx.x / (W_ / TILE)) % (H_ / TILE);
  const int b  = blockIdx.x / ((W_ / TILE) * (H_ / TILE));
  const int ox0 = bx * TILE, oy0 = by * TILE;

  if (tid < 2 * KS * KS) wsh[tid] = w[tid];

  // ---- stage SPLIT partial halo tiles into LDS ----
  for (int e = tid; e < SPLIT * 2 * TW * TW; e += 256) {
    const int pt = e / (2 * TW * TW);
    const int r2 = e - pt * (2 * TW * TW);
    const int c  = r2 / (TW * TW);
    const int r  = r2 - c * (TW * TW);
    const int ry = r / TW;
    const int rx = r - ry * TW;
    const int ih = oy0 + ry - PAD_;
    const int iw = ox0 + rx - PAD_;
    float* dst = &tile[e];
    if (ih >= 0 && ih < H_ && iw >= 0 && iw < W_) {
      const float* gp = partial + (size_t)pt * (B_ * 2 * HW)
                                + (size_t)b * (2 * HW)
                                + (size_t)c * HW + (size_t)ih * W_ + iw;
#if USE_ASYNC_LDS
      __builtin_amdgcn_global_load_async_to_lds_b32(
          (gint_t*)(void*)gp, (sint_t*)(void*)dst, 0, 0);
#else
      *dst = *gp;
#endif
    } else {
      *dst = 0.0f;   // zero padding (SAME, pad=3)
    }
  }
  WAIT_ASYNC();
  __syncthreads();

  // ---- combine partials in LDS: mean = (s0+s1)/C, max = max(m0,m1) ----
  for (int e = tid; e < 2 * TW * TW; e += 256) {
    float v = tile[e];
    if (SPLIT == 2) {
      const float v2 = tile[e + 2 * TW * TW];
      v = (e < TW * TW) ? (v + v2) : fmaxf(v, v2);
    }
    if (e < TW * TW) v *= (1.0f / (float)C_);   // mean channel
    tile[e] = v;
  }
  __syncthreads();

  // ---- 7x7 conv + sigmoid ----
  const int tx = tid & (TILE - 1);
  const int ty = tid >> 4;
  float acc = 0.0f;
#pragma unroll
  for (int c = 0; c < 2; ++c)
#pragma unroll
    for (int ki = 0; ki < KS; ++ki)
#pragma unroll
      for (int kj = 0; kj < KS; ++kj)
        acc = fmaf(wsh[c * (KS * KS) + ki * KS + kj],
                   tile[c * (TW * TW) + (ty + ki) * TW + (tx + kj)], acc);

  const float y = 1.0f / (1.0f + expf(-acc));
  out[(size_t)b * HW + (size_t)(oy0 + ty) * W_ + (ox0 + tx)] = y;
}

// ------------------------------------------------------------------
extern "C" void kernel_launch(void* const* d_in, const int* in_sizes, int n_in,
                              void* d_out, int out_size, void* d_ws, size_t ws_size,
                              hipStream_t stream) {
  (void)in_sizes; (void)n_in; (void)out_size;
  const float* x = (const float*)d_in[0];
  const float* w = (const float*)d_in[1];
  float* out     = (float*)d_out;
  float* partial = (float*)d_ws;

  const size_t part_bytes = (size_t)B_ * 2 * HW * sizeof(float);  // 3.2 MB
  const int nconv = B_ * (H_ / TILE) * (W_ / TILE);               // 1568

  if (ws_size >= 2 * part_bytes) {
    // 2-way channel split: 6272 pool waves for HBM latency hiding
    sa_pool<2><<<dim3(HW4 / POOL_BLK, B_, 2), POOL_BLK, 0, stream>>>(
        (const v4f*)x, (v4f*)partial);
    sa_conv<2><<<nconv, 256, 0, stream>>>(partial, w, out);
  } else {
    sa_pool<1><<<dim3(HW4 / POOL_BLK, B_, 1), POOL_BLK, 0, stream>>>(
        (const v4f*)x, (v4f*)partial);
    sa_conv<1><<<nconv, 256, 0, stream>>>(partial, w, out);
  }
}